// GATEncoder_61220463837358
// MI455X (gfx1250) — compile-verified
//
#include <hip/hip_runtime.h>
#include <math.h>

typedef __attribute__((ext_vector_type(2))) float v2f;
typedef __attribute__((ext_vector_type(8))) float v8f;

#define NEG_SLOPE 0.2f

// ---------------------------------------------------------------- utilities

__global__ void fill_kernel(float* __restrict__ p, float v, long long n) {
    long long i = (long long)blockIdx.x * blockDim.x + threadIdx.x;
    if (i < n) p[i] = v;
}

__device__ inline void atomicMaxFloat(float* addr, float val) {
    // valid with *addr initialized to -inf
    if (val >= 0.0f) atomicMax((int*)addr, __float_as_int(val));
    else             atomicMin((unsigned int*)addr, __float_as_uint(val));
}

// ---------------------------------------------------------------- WMMA GEMM
// C[rows x ldb] = A[rows x K] * B[K x ldb], K % 64 == 0, rows % 16 == 0.
// Block: (COLS/16) waves; each wave owns a 32x16 output tile (two 16x16
// accumulators sharing B fragments -> 2 LDS ops/WMMA, 2 independent chains).
template <int COLS>
__global__ __launch_bounds__(COLS * 2)
void gemm_wmma_kernel(const float* __restrict__ A, const float* __restrict__ B,
                      float* __restrict__ C, int nrows, int K, int ldb) {
    constexpr int KSTEP = 64;
    constexpr int LDA = KSTEP + 4;   // bank-conflict-free fragment reads
    constexpr int LDB = COLS + 4;
    constexpr int NTHREADS = COLS * 2;
    __shared__ float sA[32 * LDA];
    __shared__ float sB[KSTEP * LDB];

    const int tid   = threadIdx.x;
    const int wave  = tid >> 5;
    const int lane  = tid & 31;
    const int lrow  = lane & 15;     // M (A/D) or N (B/D) index
    const int lhalf = lane >> 4;     // K-half select for A/B fragments
    const int r0 = blockIdx.x * 32;
    const int c0 = blockIdx.y * COLS;
    const int wc = wave * 16;        // this wave's column tile inside block
    const bool tile1_valid = (r0 + 16) < nrows;

    v8f acc0 = {};
    v8f acc1 = {};

    for (int k0 = 0; k0 < K; k0 += KSTEP) {
        for (int i = tid; i < 32 * KSTEP; i += NTHREADS) {
            int row = i / KSTEP, col = i % KSTEP;
            int grow = r0 + row; if (grow >= nrows) grow = nrows - 1; // clamp OOB tail
            sA[row * LDA + col] = A[(size_t)grow * K + (k0 + col)];
        }
        for (int i = tid; i < KSTEP * COLS; i += NTHREADS) {
            int row = i / COLS, col = i % COLS;
            sB[row * LDB + col] = B[(size_t)(k0 + row) * ldb + (c0 + col)];
        }
        __syncthreads();
#pragma unroll
        for (int kk = 0; kk < KSTEP; kk += 4) {
            const int ka = kk + 2 * lhalf;
            v2f a0, a1, b;
            a0.x = sA[lrow * LDA + ka];
            a0.y = sA[lrow * LDA + ka + 1];
            a1.x = sA[(lrow + 16) * LDA + ka];
            a1.y = sA[(lrow + 16) * LDA + ka + 1];
            b.x  = sB[ka * LDB + wc + lrow];
            b.y  = sB[(ka + 1) * LDB + wc + lrow];
            acc0 = __builtin_amdgcn_wmma_f32_16x16x4_f32(
                false, a0, false, b, (short)0, acc0, false, false);
            acc1 = __builtin_amdgcn_wmma_f32_16x16x4_f32(
                false, a1, false, b, (short)0, acc1, false, false);
        }
        __syncthreads();
    }
#pragma unroll
    for (int i = 0; i < 8; ++i) {
        C[(size_t)(r0 + i + 8 * lhalf) * ldb + (c0 + wc + lrow)] = acc0[i];
    }
    if (tile1_valid) {
#pragma unroll
        for (int i = 0; i < 8; ++i) {
            C[(size_t)(r0 + 16 + i + 8 * lhalf) * ldb + (c0 + wc + lrow)] = acc1[i];
        }
    }
}

// ---------------------------------------------------------------- attention
// alpha_s[n,h] = dot(h[n,h,:64], a_src[h,:]), alpha_d likewise.
__global__ void alpha_kernel(const float* __restrict__ h,
                             const float* __restrict__ a_src,
                             const float* __restrict__ a_dst,
                             float* __restrict__ as, float* __restrict__ ad,
                             int n, int H) {
    int idx = blockIdx.x * blockDim.x + threadIdx.x;
    if (idx >= n * H) return;
    int node = idx / H, hh = idx % H;
    const float* hp  = h + (size_t)node * H * 64 + (size_t)hh * 64;
    const float* asp = a_src + hh * 64;
    const float* adp = a_dst + hh * 64;
    float s = 0.f, d = 0.f;
#pragma unroll 8
    for (int i = 0; i < 64; ++i) { float v = hp[i]; s += v * asp[i]; d += v * adp[i]; }
    as[idx] = s;
    ad[idx] = d;
}

template <int H>
__global__ void edge_max_kernel(const int* __restrict__ srcs,
                                const int* __restrict__ dsts, int E, int Etot,
                                const float* __restrict__ as,
                                const float* __restrict__ ad,
                                float* __restrict__ m) {
    int e = blockIdx.x * blockDim.x + threadIdx.x;
    if (e >= Etot) return;
    int s = (e < E) ? srcs[e] : (e - E);
    int d = (e < E) ? dsts[e] : (e - E);
#pragma unroll
    for (int h = 0; h < H; ++h) {
        float v = as[s * H + h] + ad[d * H + h];
        v = (v >= 0.f) ? v : NEG_SLOPE * v;
        atomicMaxFloat(&m[d * H + h], v);
    }
}

template <int H>
__global__ void edge_sum_kernel(const int* __restrict__ srcs,
                                const int* __restrict__ dsts, int E, int Etot,
                                const float* __restrict__ as,
                                const float* __restrict__ ad,
                                const float* __restrict__ m,
                                float* __restrict__ ex,
                                float* __restrict__ denom) {
    int e = blockIdx.x * blockDim.x + threadIdx.x;
    if (e >= Etot) return;
    int s = (e < E) ? srcs[e] : (e - E);
    int d = (e < E) ? dsts[e] : (e - E);
#pragma unroll
    for (int h = 0; h < H; ++h) {
        float v = as[s * H + h] + ad[d * H + h];
        v = (v >= 0.f) ? v : NEG_SLOPE * v;
        float xv = __expf(v - m[d * H + h]);
        ex[(size_t)e * H + h] = xv;
        atomicAdd(&denom[d * H + h], xv);
    }
}

// out[dst, f] += (ex/denom) * h[src, f] ;  F = H*64 features.
template <int H>
__global__ void agg_kernel(const int* __restrict__ srcs,
                           const int* __restrict__ dsts, int E, int Etot,
                           const float* __restrict__ hfeat,
                           const float* __restrict__ ex,
                           const float* __restrict__ denom,
                           float* __restrict__ out) {
    constexpr int F = H * 64;
    long long idx = (long long)blockIdx.x * blockDim.x + threadIdx.x;
    if (idx >= (long long)Etot * F) return;
    int e = (int)(idx / F);
    int f = (int)(idx % F);
    int s = (e < E) ? srcs[e] : (e - E);
    int d = (e < E) ? dsts[e] : (e - E);
    int h = f >> 6;
    float coeff = ex[(size_t)e * H + h] / denom[d * H + h];
    atomicAdd(&out[(size_t)d * F + f], coeff * hfeat[(size_t)s * F + f]);
}

__global__ void relu_bias_kernel(float* __restrict__ x, const float* __restrict__ b,
                                 long long n, int F) {
    long long idx = (long long)blockIdx.x * blockDim.x + threadIdx.x;
    if (idx >= n) return;
    float v = x[idx] + b[(int)(idx % F)];
    x[idx] = v > 0.f ? v : 0.f;
}

__global__ void add_bias_kernel(float* __restrict__ x, const float* __restrict__ b,
                                long long n, int F) {
    long long idx = (long long)blockIdx.x * blockDim.x + threadIdx.x;
    if (idx >= n) return;
    x[idx] += b[(int)(idx % F)];
}

// ---------------------------------------------------------------- launcher

static inline unsigned cdiv_ll(long long a, int b) { return (unsigned)((a + b - 1) / b); }

extern "C" void kernel_launch(void* const* d_in, const int* in_sizes, int n_in,
                              void* d_out, int out_size, void* d_ws, size_t ws_size,
                              hipStream_t stream) {
    const float* x      = (const float*)d_in[0];
    const int*   ei     = (const int*)d_in[1];
    const float* W1     = (const float*)d_in[2];
    const float* a_src1 = (const float*)d_in[3];
    const float* a_dst1 = (const float*)d_in[4];
    const float* b1     = (const float*)d_in[5];
    const float* W2     = (const float*)d_in[6];
    const float* a_src2 = (const float*)d_in[7];
    const float* a_dst2 = (const float*)d_in[8];
    const float* b2     = (const float*)d_in[9];

    const int E    = in_sizes[1] / 2;
    const int N    = in_sizes[0] / 256;   // IN = 256
    const int Etot = E + N;
    const int* srcs = ei;
    const int* dsts = ei + E;

    // workspace layout (floats)
    float* w = (float*)d_ws;
    size_t off = 0;
    float* h1  = w + off; off += (size_t)N * 256;   // layer-1 features (also reused for h2)
    float* x2  = w + off; off += (size_t)N * 256;   // layer-1 aggregated output / layer-2 input
    float* as1 = w + off; off += (size_t)N * 4;
    float* ad1 = w + off; off += (size_t)N * 4;
    float* m1  = w + off; off += (size_t)N * 4;
    float* dn1 = w + off; off += (size_t)N * 4;
    float* as2 = w + off; off += (size_t)N;
    float* ad2 = w + off; off += (size_t)N;
    float* m2  = w + off; off += (size_t)N;
    float* dn2 = w + off; off += (size_t)N;
    float* ex1 = w + off; off += (size_t)Etot * 4;
    float* ex2 = w + off; off += (size_t)Etot;
    float* h2  = h1;   // h1 is dead once layer-1 aggregation completes
    float* outp = (float*)d_out;

    const int T = 256;
    const unsigned rb = (unsigned)((N + 31) / 32);  // 32-row GEMM blocks

    // ---- init accumulators (graph-safe: plain kernels, re-run every call)
    fill_kernel<<<cdiv_ll((long long)N * 4, T), T, 0, stream>>>(m1, -INFINITY, (long long)N * 4);
    fill_kernel<<<cdiv_ll((long long)N * 4, T), T, 0, stream>>>(dn1, 0.f, (long long)N * 4);
    fill_kernel<<<cdiv_ll((long long)N * 256, T), T, 0, stream>>>(x2, 0.f, (long long)N * 256);
    fill_kernel<<<cdiv_ll(N, T), T, 0, stream>>>(m2, -INFINITY, N);
    fill_kernel<<<cdiv_ll(N, T), T, 0, stream>>>(dn2, 0.f, N);
    fill_kernel<<<cdiv_ll((long long)N * 64, T), T, 0, stream>>>(outp, 0.f, (long long)N * 64);

    // ---- layer 1: h1 = x @ W1  (50000x256 @ 256x256), fp32 WMMA
    gemm_wmma_kernel<128><<<dim3(rb, 2), 256, 0, stream>>>(x, W1, h1, N, 256, 256);
    alpha_kernel<<<cdiv_ll((long long)N * 4, T), T, 0, stream>>>(h1, a_src1, a_dst1, as1, ad1, N, 4);
    edge_max_kernel<4><<<cdiv_ll(Etot, T), T, 0, stream>>>(srcs, dsts, E, Etot, as1, ad1, m1);
    edge_sum_kernel<4><<<cdiv_ll(Etot, T), T, 0, stream>>>(srcs, dsts, E, Etot, as1, ad1, m1, ex1, dn1);
    agg_kernel<4><<<cdiv_ll((long long)Etot * 256, T), T, 0, stream>>>(srcs, dsts, E, Etot, h1, ex1, dn1, x2);
    relu_bias_kernel<<<cdiv_ll((long long)N * 256, T), T, 0, stream>>>(x2, b1, (long long)N * 256, 256);

    // ---- layer 2: h2 = x2 @ W2  (50000x256 @ 256x64), heads=1, mean==identity
    gemm_wmma_kernel<64><<<dim3(rb, 1), 128, 0, stream>>>(x2, W2, h2, N, 256, 64);
    alpha_kernel<<<cdiv_ll(N, T), T, 0, stream>>>(h2, a_src2, a_dst2, as2, ad2, N, 1);
    edge_max_kernel<1><<<cdiv_ll(Etot, T), T, 0, stream>>>(srcs, dsts, E, Etot, as2, ad2, m2);
    edge_sum_kernel<1><<<cdiv_ll(Etot, T), T, 0, stream>>>(srcs, dsts, E, Etot, as2, ad2, m2, ex2, dn2);
    agg_kernel<1><<<cdiv_ll((long long)Etot * 64, T), T, 0, stream>>>(srcs, dsts, E, Etot, h2, ex2, dn2, outp);
    add_bias_kernel<<<cdiv_ll((long long)N * 64, T), T, 0, stream>>>(outp, b2, (long long)N * 64, 64);

    (void)n_in; (void)out_size; (void)ws_size;
}